// SSIM_Loss_90615220011063
// MI455X (gfx1250) — compile-verified
//
#include <hip/hip_runtime.h>

typedef __attribute__((ext_vector_type(2))) float v2f;
typedef __attribute__((ext_vector_type(8))) float v8f;

#define TILE    32
#define IROWS   48
#define ISTRIDE 50   // even (b64-aligned pairs) and conflict-free across rows
#define HROWS   48
#define HSTRIDE 33   // conflict-free for the D-layout stores
#define NBLOCKS (16 * 16 * 16)
#define NVALID  4032064.0f   // 16 * 502 * 502

// 1D normalized Gaussian, size 11, sigma 1.5 (separable factor of reference window)
__device__ __constant__ float c_w[11] = {
    0.00102838f, 0.00759876f, 0.03600077f, 0.10936070f, 0.21300554f,
    0.26601173f,
    0.21300554f, 0.10936070f, 0.03600077f, 0.00759876f, 0.00102838f};

__device__ __forceinline__ float wt(int t) {
    return (t >= 0 && t <= 10) ? c_w[t] : 0.0f;
}

__device__ __forceinline__ v8f wmma4(v2f a, v2f b, v8f c) {
    return __builtin_amdgcn_wmma_f32_16x16x4_f32(false, a, false, b, (short)0, c,
                                                 false, false);
}

__global__ __launch_bounds__(128) void ssim_tile_kernel(
    const float* __restrict__ sr, const float* __restrict__ hr,
    float* __restrict__ partial) {
    __shared__ float sI[2][IROWS * ISTRIDE];   // hr, sr input window
    __shared__ float sH[5][HROWS * HSTRIDE];   // horizontally-convolved signals
    __shared__ float sRed[4];

    const int tid  = threadIdx.x;
    const int wave = tid >> 5;
    const int lane = tid & 31;
    const int p    = lane & 15;           // row (A) / col (B,D) index within 16
    const int half = lane >> 4;           // 0 or 1
    const int k0   = half << 1;           // K sub-offset {0,2}

    const int img = blockIdx.z;
    const int gr0 = blockIdx.y * TILE;
    const int gc0 = blockIdx.x * TILE;
    const long base = (long)img * 3 * 512 * 512;  // channel 0 of NCHW

    // ---- Stage 0: load clamped 48x48 window of hr and sr into LDS ----
    for (int idx = tid; idx < IROWS * IROWS; idx += 128) {
        int rr = idx / IROWS;
        int cc = idx - rr * IROWS;
        int gr = min(gr0 + rr, 511);
        int gc = min(gc0 + cc, 511);
        long off = base + (long)gr * 512 + gc;
        sI[0][rr * ISTRIDE + cc] = hr[off];
        sI[1][rr * ISTRIDE + cc] = sr[off];
    }

    // Toeplitz fragments: tz[k] serves as B in the horizontal pass and as A in
    // the vertical pass (identical per-lane values for the f32 WMMA layouts).
    v2f tz[7];
#pragma unroll
    for (int k = 0; k < 7; ++k) {
        int j0  = 4 * k + k0;
        tz[k].x = wt(j0 - p);
        tz[k].y = wt(j0 + 1 - p);
    }
    __syncthreads();

    // ---- Stage 1: horizontal pass, 6 subtile ops over 4 waves ----
    // rows {0,16,32} x output-col-groups {0,16}; H[r][n] = sum_j I[r][j]*w[j-n]
    for (int op = wave; op < 6; op += 4) {
        const int r0  = (op >> 1) * 16;
        const int cc0 = (op & 1) * 16;
        v8f a0 = {}, a1 = {}, a2 = {}, a3 = {}, a4 = {};
#pragma unroll
        for (int k = 0; k < 7; ++k) {
            int col = cc0 + 4 * k + k0;
            const float* ph = &sI[0][(r0 + p) * ISTRIDE + col];
            const float* ps = &sI[1][(r0 + p) * ISTRIDE + col];
            v2f h2, s2;
            h2.x = ph[0]; h2.y = ph[1];
            s2.x = ps[0]; s2.y = ps[1];
            v2f hh = h2 * h2, ss = s2 * s2, hs = h2 * s2;
            a0 = wmma4(h2, tz[k], a0);
            a1 = wmma4(s2, tz[k], a1);
            a2 = wmma4(hh, tz[k], a2);
            a3 = wmma4(ss, tz[k], a3);
            a4 = wmma4(hs, tz[k], a4);
        }
#pragma unroll
        for (int v = 0; v < 8; ++v) {
            int row = r0 + v + (half << 3);
            int col = cc0 + p;
            int o   = row * HSTRIDE + col;
            sH[0][o] = a0[v];
            sH[1][o] = a1[v];
            sH[2][o] = a2[v];
            sH[3][o] = a3[v];
            sH[4][o] = a4[v];
        }
    }
    __syncthreads();

    // ---- Stage 2: vertical pass, one 16x16 output subtile per wave ----
    const int r0 = (wave >> 1) * 16;
    const int c0 = (wave & 1) * 16;
    v8f m1 = {}, m2 = {}, qhh = {}, qss = {}, qhs = {};
#pragma unroll
    for (int k = 0; k < 7; ++k) {
        int j  = r0 + 4 * k + k0;
        int ci = c0 + p;
        v2f b;
        b.x = sH[0][j * HSTRIDE + ci]; b.y = sH[0][(j + 1) * HSTRIDE + ci];
        m1 = wmma4(tz[k], b, m1);
        b.x = sH[1][j * HSTRIDE + ci]; b.y = sH[1][(j + 1) * HSTRIDE + ci];
        m2 = wmma4(tz[k], b, m2);
        b.x = sH[2][j * HSTRIDE + ci]; b.y = sH[2][(j + 1) * HSTRIDE + ci];
        qhh = wmma4(tz[k], b, qhh);
        b.x = sH[3][j * HSTRIDE + ci]; b.y = sH[3][(j + 1) * HSTRIDE + ci];
        qss = wmma4(tz[k], b, qss);
        b.x = sH[4][j * HSTRIDE + ci]; b.y = sH[4][(j + 1) * HSTRIDE + ci];
        qhs = wmma4(tz[k], b, qhs);
    }

    // ---- Stage 3: pointwise SSIM + masked sum ----
    const float C1 = 1e-4f, C2 = 9e-4f;
    float lsum = 0.0f;
#pragma unroll
    for (int v = 0; v < 8; ++v) {
        int gr = gr0 + r0 + v + (half << 3);
        int gc = gc0 + c0 + p;
        float mu1 = m1[v], mu2 = m2[v];
        float mu1s = mu1 * mu1, mu2s = mu2 * mu2, m12 = mu1 * mu2;
        float s1  = qhh[v] - mu1s;
        float s2v = qss[v] - mu2s;
        float s12 = qhs[v] - m12;
        float num = (2.0f * m12 + C1) * (2.0f * s12 + C2);
        float den = (mu1s + mu2s + C1) * (s1 + s2v + C2);
        float val = num / den;
        if (gr < 502 && gc < 502) lsum += val;
    }
#pragma unroll
    for (int off = 16; off > 0; off >>= 1) lsum += __shfl_down(lsum, off, 32);
    if (lane == 0) sRed[wave] = lsum;
    __syncthreads();
    if (tid == 0) {
        float t = sRed[0] + sRed[1] + sRed[2] + sRed[3];
        partial[(blockIdx.z * gridDim.y + blockIdx.y) * gridDim.x + blockIdx.x] = t;
    }
}

__global__ __launch_bounds__(256) void ssim_reduce_kernel(
    const float* __restrict__ partial, float* __restrict__ out, int n) {
    __shared__ float s[256];
    float a = 0.0f;
    for (int i = threadIdx.x; i < n; i += 256) a += partial[i];
    s[threadIdx.x] = a;
    __syncthreads();
    for (int st = 128; st > 0; st >>= 1) {
        if ((int)threadIdx.x < st) s[threadIdx.x] += s[threadIdx.x + st];
        __syncthreads();
    }
    if (threadIdx.x == 0) out[0] = s[0] * (1.0f / NVALID);
}

extern "C" void kernel_launch(void* const* d_in, const int* in_sizes, int n_in,
                              void* d_out, int out_size, void* d_ws,
                              size_t ws_size, hipStream_t stream) {
    const float* sr = (const float*)d_in[0];  // sr_image
    const float* hr = (const float*)d_in[1];  // hr_image
    float* out      = (float*)d_out;
    float* partial  = (float*)d_ws;           // 4096 floats

    dim3 grid(16, 16, 16);  // x: col tiles, y: row tiles, z: images
    ssim_tile_kernel<<<grid, 128, 0, stream>>>(sr, hr, partial);
    ssim_reduce_kernel<<<1, 256, 0, stream>>>(partial, out, NBLOCKS);
}